// GridEncoder_74234214744826
// MI455X (gfx1250) — compile-verified
//
#include <hip/hip_runtime.h>

// ---------------------------------------------------------------------------
// GridEncoder (instant-NGP hashed triplane): B=2M points, 3 levels, C=2.
// All levels hash with primes (1, 2654435761, 805459861), hmap = 2^19.
// Table layout in d_in[0]: triplane [3, 2, 1024, 512]  (plane, chan, h, w)
//   emb[plane*HW + local, c] == triplane[plane, c, local/512, local%512]
// Strategy: pre-transpose table into interleaved float2 rows in d_ws so each
// corner gather is ONE global_load_b64 (L2-resident, 192MB L2 >> 52MB set).
// ---------------------------------------------------------------------------

#define GE_HW      524288u          // 1024*512 entries per plane
#define GE_NPLANE  3
#define GE_MASK    0x7FFFFu        // hmap = 2^19 (power of two -> AND)
#define GE_P1      2654435761u
#define GE_P2      805459861u

// ---- pre-pass: [3,2,HW] -> interleaved [3*HW] float2 (coalesced both ways)
__global__ __launch_bounds__(256)
void ge_transpose_kernel(const float* __restrict__ tp, float2* __restrict__ emb) {
    unsigned i = blockIdx.x * blockDim.x + threadIdx.x;      // 0 .. 3*HW-1
    if (i >= GE_NPLANE * GE_HW) return;
    unsigned plane = i >> 19;                                // / HW
    unsigned local = i & (GE_HW - 1u);
    const float* base = tp + (size_t)plane * 2u * GE_HW;
    float2 v;
    v.x = base[local];            // channel 0 slab (coalesced)
    v.y = base[GE_HW + local];    // channel 1 slab (coalesced)
    emb[i] = v;
}

// ---- main encode kernel --------------------------------------------------
template <bool INTERLEAVED>
__global__ __launch_bounds__(256)
void ge_encode_kernel(const float*  __restrict__ tp,    // native layout (fallback)
                      const float2* __restrict__ emb,   // interleaved table
                      const float*  __restrict__ xin,   // [B,3] in [-1,1]
                      float2*       __restrict__ out,   // [B] float2
                      int nb)
{
    int tid    = blockIdx.x * blockDim.x + threadIdx.x;
    int stride = gridDim.x * blockDim.x;

    for (int b = tid; b < nb; b += stride) {
        // prefetch next chunk of coordinates (gfx1250 global_prefetch_b8)
        if (b + stride < nb)
            __builtin_prefetch(xin + 3 * (b + stride), 0, 3);

        // per-lane 12B coalesced coordinate load
        float cx = xin[3 * b + 0];
        float cy = xin[3 * b + 1];
        float cz = xin[3 * b + 2];
        // map [-1,1] -> [0,1]
        float x01 = __fmul_rn(__fadd_rn(cx, 1.0f), 0.5f);
        float y01 = __fmul_rn(__fadd_rn(cy, 1.0f), 0.5f);
        float z01 = __fmul_rn(__fadd_rn(cz, 1.0f), 0.5f);

        unsigned idxs[24];
        float    wgt[24];

        #pragma unroll
        for (int l = 0; l < 3; ++l) {
            const float scale = (float)((128 << l) - 1);   // 127, 255, 511
            // pos = x01*scale + 0.5 with the SAME rounding as the reference
            // (block fp-contraction so floor() picks the identical cell)
            float px = __fadd_rn(__fmul_rn(x01, scale), 0.5f);
            float py = __fadd_rn(__fmul_rn(y01, scale), 0.5f);
            float pz = __fadd_rn(__fmul_rn(z01, scale), 0.5f);
            float gx = floorf(px), gy = floorf(py), gz = floorf(pz);
            float fx = px - gx,    fy = py - gy,    fz = pz - gz;

            // hash terms; (v+1)*p == v*p + p in uint32 wraparound arithmetic
            unsigned hx0 = (unsigned)gx;               // prime 1
            unsigned hx1 = hx0 + 1u;
            unsigned hy0 = (unsigned)gy * GE_P1;
            unsigned hy1 = hy0 + GE_P1;
            unsigned hz0 = (unsigned)gz * GE_P2;
            unsigned hz1 = hz0 + GE_P2;

            float wx0 = 1.0f - fx, wy0 = 1.0f - fy, wz0 = 1.0f - fz;
            unsigned lbase = (unsigned)l * GE_HW;

            #pragma unroll
            for (int c = 0; c < 8; ++c) {
                unsigned h = ((c & 1) ? hx1 : hx0)
                           ^ ((c & 2) ? hy1 : hy0)
                           ^ ((c & 4) ? hz1 : hz0);
                idxs[l * 8 + c] = lbase + (h & GE_MASK);
                wgt [l * 8 + c] = ((c & 1) ? fx : wx0)
                                * ((c & 2) ? fy : wy0)
                                * ((c & 4) ? fz : wz0);
            }
        }

        // issue ALL 24 gathers up front -> long vmem clause, single wait
        float2 f[24];
        if (INTERLEAVED) {
            #pragma unroll
            for (int k = 0; k < 24; ++k)
                f[k] = emb[idxs[k]];                    // one global_load_b64 each
        } else {
            #pragma unroll
            for (int l = 0; l < 3; ++l) {
                const float* b0 = tp + (size_t)l * 2u * GE_HW;  // chan0 slab
                const float* b1 = b0 + GE_HW;                   // chan1 slab
                #pragma unroll
                for (int c = 0; c < 8; ++c) {
                    unsigned li = idxs[l * 8 + c] - (unsigned)l * GE_HW;
                    f[l * 8 + c].x = b0[li];
                    f[l * 8 + c].y = b1[li];
                }
            }
        }

        float2 acc = make_float2(0.0f, 0.0f);
        #pragma unroll
        for (int k = 0; k < 24; ++k) {
            acc.x = fmaf(wgt[k], f[k].x, acc.x);
            acc.y = fmaf(wgt[k], f[k].y, acc.y);
        }
        out[b] = acc;                                  // coalesced b64 store
    }
}

// ---------------------------------------------------------------------------
extern "C" void kernel_launch(void* const* d_in, const int* in_sizes, int n_in,
                              void* d_out, int out_size, void* d_ws, size_t ws_size,
                              hipStream_t stream) {
    const float* tp  = (const float*)d_in[0];   // triplane [3,2,1024,512]
    const float* xin = (const float*)d_in[1];   // inputs   [B,3]
    float2*      out = (float2*)d_out;          // [B,2] f32
    int nb = in_sizes[1] / 3;

    const size_t needed = (size_t)GE_NPLANE * GE_HW * sizeof(float2); // 12.6 MB
    const int threads = 256;                       // 8 wave32 waves / block
    int blocks = (nb + threads * 2 - 1) / (threads * 2);  // 2 points / thread
    if (blocks < 1) blocks = 1;

    if (ws_size >= needed) {
        float2* emb = (float2*)d_ws;
        int tblocks = (GE_NPLANE * GE_HW + threads - 1) / threads;
        ge_transpose_kernel<<<tblocks, threads, 0, stream>>>(tp, emb);
        ge_encode_kernel<true><<<blocks, threads, 0, stream>>>(tp, emb, xin, out, nb);
    } else {
        ge_encode_kernel<false><<<blocks, threads, 0, stream>>>(tp, nullptr, xin, out, nb);
    }
}